// SPAIRGlimpseZPresMLP_15470472200210
// MI455X (gfx1250) — compile-verified
//
#include <hip/hip_runtime.h>
#include <hip/hip_bf16.h>

typedef __attribute__((ext_vector_type(2))) float v2f;
typedef __attribute__((ext_vector_type(8))) float v8f;

static constexpr int CH = 8;  // items per thread in segment passes

// ---- ordered-uint encoding for float atomicMax (handles negatives) ----
__device__ __forceinline__ unsigned enc_f32(float f) {
  unsigned u = __float_as_uint(f);
  return (u & 0x80000000u) ? ~u : (u | 0x80000000u);
}
__device__ __forceinline__ float dec_f32(unsigned u) {
  return (u & 0x80000000u) ? __uint_as_float(u & 0x7FFFFFFFu)
                           : __uint_as_float(~u);
}
// encoded(-inf): bits(-inf)=0xFF800000 -> ~ = 0x007FFFFF
#define ENC_NEG_INF 0x007FFFFFu

// ---------------- K0: init workspace ----------------
__global__ void k_init(unsigned* __restrict__ segmax, float* __restrict__ segsum,
                       float* __restrict__ center, int G) {
  int i = blockIdx.x * blockDim.x + threadIdx.x;
  if (i < G) {
    segmax[i] = ENC_NEG_INF;
    segsum[i] = 0.0f;
    center[3 * i + 0] = 0.0f;
    center[3 * i + 1] = 0.0f;
    center[3 * i + 2] = 0.0f;
  }
}

// ---------------- K1: segment max (sorted index, run-compressed atomics) ----
__global__ void k_segmax(const float* __restrict__ lm, const int* __restrict__ idx,
                         unsigned* __restrict__ segmax, int M) {
  int t = blockIdx.x * blockDim.x + threadIdx.x;
  int base = t * CH;
  if (base >= M) return;
  int end = (base + CH <= M) ? CH : (M - base);
  int cur = idx[base];
  float mx = lm[base];
  for (int j = 1; j < end; ++j) {
    int g = idx[base + j];
    float v = lm[base + j];
    if (g == cur) {
      mx = fmaxf(mx, v);
    } else {
      atomicMax(&segmax[cur], enc_f32(mx));
      cur = g;
      mx = v;
    }
  }
  atomicMax(&segmax[cur], enc_f32(mx));
}

// ---------------- K2: segment sum of exp(shifted) and exp-weighted pos ------
__global__ void k_segacc(const float* __restrict__ lm, const int* __restrict__ idx,
                         const float* __restrict__ pos,
                         const unsigned* __restrict__ segmax,
                         float* __restrict__ segsum, float* __restrict__ center,
                         int M) {
  int t = blockIdx.x * blockDim.x + threadIdx.x;
  int base = t * CH;
  if (base >= M) return;
  int end = (base + CH <= M) ? CH : (M - base);

  int cur = idx[base];
  float smax = dec_f32(segmax[cur]);
  float se = 0.0f, cx = 0.0f, cy = 0.0f, cz = 0.0f;
  for (int j = 0; j < end; ++j) {
    int i = base + j;
    int g = idx[i];
    if (g != cur) {
      atomicAdd(&segsum[cur], se);
      atomicAdd(&center[3 * cur + 0], cx);
      atomicAdd(&center[3 * cur + 1], cy);
      atomicAdd(&center[3 * cur + 2], cz);
      cur = g;
      smax = dec_f32(segmax[cur]);
      se = cx = cy = cz = 0.0f;
    }
    float e = expf(lm[i] - smax);
    se += e;
    cx += e * pos[3 * i + 0];
    cy += e * pos[3 * i + 1];
    cz += e * pos[3 * i + 2];
  }
  atomicAdd(&segsum[cur], se);
  atomicAdd(&center[3 * cur + 0], cx);
  atomicAdd(&center[3 * cur + 1], cy);
  atomicAdd(&center[3 * cur + 2], cz);
}

// ---------------- K3: member_center = (sum e*pos) / (sum e) -----------------
__global__ void k_center(const float* __restrict__ segsum,
                         const float* __restrict__ center, float* __restrict__ out,
                         int G) {
  int i = blockIdx.x * blockDim.x + threadIdx.x;
  if (i >= G) return;
  float s = segsum[i];
  float inv = (s > 0.0f) ? (1.0f / s) : 0.0f;  // empty segment -> 0
  out[2 * G + 3 * i + 0] = center[3 * i + 0] * inv;
  out[2 * G + 3 * i + 1] = center[3 * i + 1] * inv;
  out[2 * G + 3 * i + 2] = center[3 * i + 2] * inv;
}

// temperature arrives as 1-elem array of unknown encoding (python int 1 or f32)
__device__ __forceinline__ float decode_temp(unsigned u) {
  if (u == 0u) return 1.0f;
  if (u < 0x00800000u) return (float)(int)u;  // small int bit pattern
  return __uint_as_float(u);                  // proper float bit pattern
}

// ---------------- K4: GEMV via V_WMMA_F32_16X16X4_F32 + epilogue ------------
// One wave handles 16 rows. A tile layout (ISA 16x4 f32): lane<16 holds
// A[m][k],A[m][k+1]; lane>=16 holds A[m][k+2],A[m][k+3]. B 4x16 mirrors:
// v0/v1 = rows K(=k+0,k+1) for lanes 0-15, rows K(=k+2,k+3) for lanes 16-31.
// B columns all replicate w -> D[m][n] independent of n.
__global__ void __launch_bounds__(256) k_gemv(const float* __restrict__ feat,
                                              const float* __restrict__ w,
                                              const float* __restrict__ bptr,
                                              const float* __restrict__ noise,
                                              const unsigned* __restrict__ traw,
                                              float* __restrict__ out, int G, int K) {
  __shared__ float ws[1024];
  for (int t = threadIdx.x; t < K; t += blockDim.x) ws[t] = w[t];
  __syncthreads();

  int lane = threadIdx.x & 31;
  int wave = blockIdx.x * (blockDim.x >> 5) + (threadIdx.x >> 5);
  int r0 = wave * 16;
  if (r0 >= G) return;

  int half = lane >> 4;   // 0: K pair (k,k+1), 1: K pair (k+2,k+3)
  int m = lane & 15;      // A row within tile
  const float* arow = feat + (size_t)(r0 + m) * K + 2 * half;

  v8f c = {0.f, 0.f, 0.f, 0.f, 0.f, 0.f, 0.f, 0.f};
#pragma unroll 4
  for (int k = 0; k < K; k += 4) {
    v2f a = *(const v2f*)(arow + k);
    v2f bb = *(const v2f*)&ws[k + 2 * half];
    c = __builtin_amdgcn_wmma_f32_16x16x4_f32(false, a, false, bb, (short)0, c,
                                              false, false);
  }

  float bias = bptr[0];
  float temp = decode_temp(traw[0]);

  // D[j][*] (rows r0+j) sits in VGPR j for lanes 0-15, rows r0+8+j for 16-31.
  // Lane j / j+16 reads c[j] = dot(row r0 + j + 8*half).
#pragma unroll
  for (int j = 0; j < 8; ++j) {
    if (m == j) {
      int g = r0 + j + 8 * half;
      float dot = c[j] + bias;
      float logit = 8.8f * tanhf(dot);
      float n = noise[g];
      float logistic = logf(n) - log1pf(-n);
      float x = (logit + logistic) / temp;
      // stable log_sigmoid
      float lz = fminf(x, 0.0f) - log1pf(expf(-fabsf(x)));
      out[g] = lz;          // log_z_pres
      out[G + g] = logit;   // logit_pres
    }
  }
}

extern "C" void kernel_launch(void* const* d_in, const int* in_sizes, int n_in,
                              void* d_out, int out_size, void* d_ws, size_t ws_size,
                              hipStream_t stream) {
  const float* feat = (const float*)d_in[0];      // (G,256)
  const float* pos = (const float*)d_in[1];       // (M,3)
  const float* lm = (const float*)d_in[2];        // (M,1)
  const int* idx = (const int*)d_in[3];           // (M,) sorted
  const float* w = (const float*)d_in[4];         // (256,1)
  const float* b = (const float*)d_in[5];         // (1,)
  const float* noise = (const float*)d_in[6];     // (G,)
  const unsigned* traw = (const unsigned*)d_in[7];// temperature (1,)
  float* out = (float*)d_out;                     // [G lz | G logit | 3G center]

  int G = in_sizes[6];
  int M = in_sizes[3];
  int K = in_sizes[0] / G;  // 256

  unsigned* segmax = (unsigned*)d_ws;             // G uints
  float* segsum = (float*)d_ws + G;               // G floats
  float* center = (float*)d_ws + 2 * (size_t)G;   // 3G floats

  int gblk = (G + 255) / 256;
  k_init<<<gblk, 256, 0, stream>>>(segmax, segsum, center, G);

  int nthreads = (M + CH - 1) / CH;
  int mblk = (nthreads + 255) / 256;
  k_segmax<<<mblk, 256, 0, stream>>>(lm, idx, segmax, M);
  k_segacc<<<mblk, 256, 0, stream>>>(lm, idx, pos, segmax, segsum, center, M);
  k_center<<<gblk, 256, 0, stream>>>(segsum, center, out, G);

  int waves = (G + 15) / 16;
  int wblk = (waves + 7) / 8;  // 8 waves (256 threads) per block
  k_gemv<<<wblk, 256, 0, stream>>>(feat, w, b, noise, traw, out, G, K);
}